// NodeEmbDecoder_66992899883305
// MI455X (gfx1250) — compile-verified
//
#include <hip/hip_runtime.h>
#include <math.h>

// Problem constants (match reference)
#define Hh   4
#define Cc   32
#define Dd   128
#define OUTF 64
#define Ll   3
#define EPSV 1e-5f

typedef __attribute__((ext_vector_type(2))) float v2f;
typedef __attribute__((ext_vector_type(8))) float v8f;

// ---- monotone float <-> uint mapping for atomicMax on floats ----
__device__ __forceinline__ unsigned fenc(float f) {
  unsigned u = __float_as_uint(f);
  return (u & 0x80000000u) ? ~u : (u | 0x80000000u);
}
__device__ __forceinline__ float fdec(unsigned u) {
  return (u & 0x80000000u) ? __uint_as_float(u & 0x7FFFFFFFu)
                           : __uint_as_float(~u);
}
#define ENC_NEG_INF 0x007FFFFFu  // fenc(-inf)

// =====================================================================
// WMMA GEMM: Cout[M,NC] = A[M,128] @ W[128,NC] + bias[NC]   (fp32 WMMA)
// W staged once per block into LDS in K-pair-interleaved layout:
//   sW[((k>>1)*NC + c)*2 + (k&1)] = W[k][c]
// so each lane's B fragment (W[k][col], W[k+1][col]) is one aligned
// ds_load_b64 directly into the even VGPR pair WMMA consumes.
// One wave -> 16 rows x NC cols. K stepped by 4 (V_WMMA_F32_16X16X4_F32).
// =====================================================================
template <int NC>
__global__ __launch_bounds__(256) void wmma_gemm_bias(
    const float* __restrict__ A, const float* __restrict__ W,
    const float* __restrict__ bias, float* __restrict__ Cout, int M) {
  __shared__ float sW[Dd * NC];

  // cooperative stage of W[128,NC] -> pair-interleaved LDS
  for (int i = threadIdx.x; i < Dd * NC / 4; i += 256) {
    const int k = i / (NC / 4);
    const int c0 = (i % (NC / 4)) * 4;
    float4 w = ((const float4*)W)[i];
    float* dst = sW + ((size_t)(k >> 1) * NC) * 2 + (k & 1);
    dst[(c0 + 0) * 2] = w.x;
    dst[(c0 + 1) * 2] = w.y;
    dst[(c0 + 2) * 2] = w.z;
    dst[(c0 + 3) * 2] = w.w;
  }
  __syncthreads();

  const int lane = threadIdx.x & 31;
  const int wave = threadIdx.x >> 5;
  const int row_tile = blockIdx.x * 8 + wave;
  const int row0 = row_tile * 16;
  if (row0 < M) {  // wave-uniform: EXEC stays all-ones inside
    const int m = lane & 15;            // A row within tile (both lane halves)
    const int n = lane & 15;            // B/D column within 16-col tile
    const int khalf = (lane >> 4) * 2;  // lanes16-31 take K+2,K+3

    v8f acc[NC / 16];
#pragma unroll
    for (int j = 0; j < NC / 16; ++j) {
      const float bj = bias[j * 16 + n];
#pragma unroll
      for (int r = 0; r < 8; ++r) acc[j][r] = bj;
    }

    const float* arow = A + (size_t)(row0 + m) * Dd;
#pragma unroll 4
    for (int ks = 0; ks < Dd / 4; ++ks) {
      const int k0 = ks * 4;
      // A frag 16x4: lane holds A[m][k0+khalf], A[m][k0+khalf+1]
      v2f a = *(const v2f*)(arow + k0 + khalf);
      // pair row for this lane's K half
      const float* prow = sW + ((size_t)((k0 + khalf) >> 1) * NC) * 2;
#pragma unroll
      for (int j = 0; j < NC / 16; ++j) {
        // single aligned b64: {W[k0+khalf][col], W[k0+khalf+1][col]}
        v2f b = *(const v2f*)(prow + (j * 16 + n) * 2);
        acc[j] = __builtin_amdgcn_wmma_f32_16x16x4_f32(
            false, a, false, b, (short)0, acc[j], false, false);
      }
    }

    // D layout: VGPR r -> M = r (lanes 0-15) / r+8 (lanes 16-31), N = lane&15
    const int rofs = (lane >> 4) << 3;
#pragma unroll
    for (int j = 0; j < NC / 16; ++j) {
#pragma unroll
      for (int r = 0; r < 8; ++r) {
        Cout[(size_t)(row0 + r + rofs) * NC + j * 16 + n] = acc[j][r];
      }
    }
  }
}

// ===================== per-layer init (agg, denom, segmax) ============
__global__ void init_layer_kernel(float* __restrict__ agg,
                                  float* __restrict__ denom,
                                  unsigned* __restrict__ m_enc, int Nn) {
  int t = blockIdx.x * blockDim.x + threadIdx.x;
  if (t < Nn * Dd) agg[t] = 0.0f;
  if (t < Nn * Hh) {
    denom[t] = 0.0f;
    m_enc[t] = ENC_NEG_INF;
  }
}

// == edge pass 1: logits (stored for pass 2) + segment-max via atomicMax ==
__global__ __launch_bounds__(256) void edge_max_kernel(
    const float* __restrict__ q, const float* __restrict__ k,
    const int* __restrict__ srcI, const int* __restrict__ dstI,
    unsigned* __restrict__ m_enc, float* __restrict__ score_buf, int Ee) {
  int t = blockIdx.x * blockDim.x + threadIdx.x;
  if (t >= Ee * Hh) return;
  int e = t >> 2, h = t & 3;
  int s = srcI[e], d = dstI[e];
  const float4* qp = (const float4*)(q + (size_t)d * Dd + h * Cc);
  const float4* kp = (const float4*)(k + (size_t)s * Dd + h * Cc);
  float acc = 0.f;
#pragma unroll
  for (int i = 0; i < Cc / 4; ++i) {
    float4 a = qp[i], b = kp[i];
    acc += a.x * b.x + a.y * b.y + a.z * b.z + a.w * b.w;
  }
  float score = acc * 0.17677669529663687f;  // 1/sqrt(32)
  score_buf[t] = score;
  atomicMax(m_enc + (size_t)d * Hh + h, fenc(score));
}

// ===== edge pass 2: denom += exp, agg += exp * v[src] (unnormalized) ==
__global__ __launch_bounds__(256) void edge_accum_kernel(
    const float* __restrict__ v, const int* __restrict__ srcI,
    const int* __restrict__ dstI, const unsigned* __restrict__ m_enc,
    const float* __restrict__ score_buf, float* __restrict__ denom,
    float* __restrict__ agg, int Ee) {
  int t = blockIdx.x * blockDim.x + threadIdx.x;
  if (t >= Ee * Hh) return;
  int e = t >> 2, h = t & 3;
  int s = srcI[e], d = dstI[e];
  float mx = fdec(m_enc[(size_t)d * Hh + h]);
  float ex = __expf(score_buf[t] - mx);
  atomicAdd(denom + (size_t)d * Hh + h, ex);
  const float4* vp = (const float4*)(v + (size_t)s * Dd + h * Cc);
  float* ap = agg + (size_t)d * Dd + h * Cc;
#pragma unroll
  for (int i = 0; i < Cc / 4; ++i) {
    float4 vv = vp[i];
    atomicAdd(ap + 4 * i + 0, ex * vv.x);
    atomicAdd(ap + 4 * i + 1, ex * vv.y);
    atomicAdd(ap + 4 * i + 2, ex * vv.z);
    atomicAdd(ap + 4 * i + 3, ex * vv.w);
  }
}

// ===== node pass: pre = agg/denom + skip  (isolated node -> skip only) =
__global__ void attn_combine_kernel(const float* __restrict__ agg,
                                    const float* __restrict__ denom,
                                    const float* __restrict__ sbuf,
                                    float* __restrict__ pre, int Nn) {
  int t = blockIdx.x * blockDim.x + threadIdx.x;
  if (t >= Nn * Dd) return;
  int nrow = t >> 7;  // /128
  int f = t & (Dd - 1);
  int h = f >> 5;  // /32
  float dnm = denom[(size_t)nrow * Hh + h];
  float inv = dnm > 0.f ? 1.f / dnm : 0.f;
  pre[t] = agg[t] * inv + sbuf[t];
}

// ===================== batch-norm statistics ==========================
__global__ void zero_stats_kernel(float* __restrict__ stats) {
  stats[threadIdx.x] = 0.f;  // blockDim.x == 256 covers sum+sumsq slots
}

template <int NC>
__global__ void colstats_kernel(const float* __restrict__ pre,
                                float* __restrict__ colsum,
                                float* __restrict__ colsumsq, int Nn) {
  int f = threadIdx.x;  // blockDim.x == NC
  size_t row0 = (size_t)blockIdx.x * 64;
  float s = 0.f, q = 0.f;
  for (int r = 0; r < 64; ++r) {
    size_t nrow = row0 + r;
    if (nrow >= (size_t)Nn) break;
    float val = pre[nrow * NC + f];
    s += val;
    q += val * val;
  }
  atomicAdd(colsum + f, s);
  atomicAdd(colsumsq + f, q);
}

template <int NC>
__global__ void bn_finalize_kernel(const float* __restrict__ colsum,
                                   const float* __restrict__ colsumsq,
                                   const float* __restrict__ g,
                                   const float* __restrict__ b,
                                   float* __restrict__ scale,
                                   float* __restrict__ shift, int Nn) {
  int f = threadIdx.x;
  if (f >= NC) return;
  float invN = 1.f / (float)Nn;
  float mean = colsum[f] * invN;
  float var = colsumsq[f] * invN - mean * mean;
  float inv = rsqrtf(var + EPSV);
  float sc = g[f] * inv;
  scale[f] = sc;
  shift[f] = b[f] - mean * sc;
}

template <int NC>
__global__ void bn_apply_relu_kernel(const float* __restrict__ pre,
                                     const float* __restrict__ scale,
                                     const float* __restrict__ shift,
                                     float* __restrict__ outp, long total) {
  long t = (long)blockIdx.x * blockDim.x + threadIdx.x;
  if (t >= total) return;
  int f = (int)(t & (NC - 1));
  float v = pre[t] * scale[f] + shift[f];
  outp[t] = v > 0.f ? v : 0.f;
}

// =====================================================================
extern "C" void kernel_launch(void* const* d_in, const int* in_sizes, int n_in,
                              void* d_out, int out_size, void* d_ws,
                              size_t ws_size, hipStream_t stream) {
  const float* x     = (const float*)d_in[0];
  const int*   ei    = (const int*)d_in[1];
  const float* Wq    = (const float*)d_in[2];
  const float* bq    = (const float*)d_in[3];
  const float* Wk    = (const float*)d_in[4];
  const float* bk    = (const float*)d_in[5];
  const float* Wv    = (const float*)d_in[6];
  const float* bv    = (const float*)d_in[7];
  const float* Ws    = (const float*)d_in[8];
  const float* bs    = (const float*)d_in[9];
  const float* bn_g  = (const float*)d_in[10];
  const float* bn_b  = (const float*)d_in[11];
  const float* W_out = (const float*)d_in[12];
  const float* b_out = (const float*)d_in[13];
  const float* bno_g = (const float*)d_in[14];
  const float* bno_b = (const float*)d_in[15];

  const int Nn = in_sizes[0] / Dd;
  const int Ee = in_sizes[1] / 2;
  const int* srcI = ei;
  const int* dstI = ei + Ee;

  // ---- workspace layout (floats) ----
  float* ws = (float*)d_ws;
  const size_t nd = (size_t)Nn * Dd;
  float*    buf_h   = ws;                                          // [N,128]
  float*    buf_q   = ws + nd;                                     // q / pre
  float*    buf_k   = ws + 2 * nd;                                 // k / head out
  float*    buf_v   = ws + 3 * nd;                                 // v
  float*    buf_s   = ws + 4 * nd;                                 // skip term
  float*    buf_agg = ws + 5 * nd;                                 // weighted sum
  float*    denom   = ws + 6 * nd;                                 // [N,4]
  unsigned* m_enc   = (unsigned*)(ws + 6 * nd + (size_t)Nn * Hh);  // [N,4]
  float*    stats   = ws + 6 * nd + 2 * (size_t)Nn * Hh;           // 512 floats
  float*    buf_score = stats + 512;                               // [E,4]
  float* colsum   = stats;
  float* colsumsq = stats + 128;
  float* scale    = stats + 256;
  float* shift    = stats + 384;

  const dim3 gemmGrid((Nn / 16 + 7) / 8);
  const int edgeBlocks = (Ee * Hh + 255) / 256;
  const int nodeBlocks = ((int)nd + 255) / 256;
  const int statBlocks = (Nn + 63) / 64;

  const float* hin = x;
  for (int l = 0; l < Ll; ++l) {
    const size_t wofs = (size_t)l * Dd * Dd;
    const size_t bofs = (size_t)l * Dd;
    wmma_gemm_bias<Dd><<<gemmGrid, 256, 0, stream>>>(hin, Wq + wofs, bq + bofs, buf_q, Nn);
    wmma_gemm_bias<Dd><<<gemmGrid, 256, 0, stream>>>(hin, Wk + wofs, bk + bofs, buf_k, Nn);
    wmma_gemm_bias<Dd><<<gemmGrid, 256, 0, stream>>>(hin, Wv + wofs, bv + bofs, buf_v, Nn);
    wmma_gemm_bias<Dd><<<gemmGrid, 256, 0, stream>>>(hin, Ws + wofs, bs + bofs, buf_s, Nn);

    init_layer_kernel<<<nodeBlocks, 256, 0, stream>>>(buf_agg, denom, m_enc, Nn);
    edge_max_kernel<<<edgeBlocks, 256, 0, stream>>>(buf_q, buf_k, srcI, dstI,
                                                    m_enc, buf_score, Ee);
    edge_accum_kernel<<<edgeBlocks, 256, 0, stream>>>(buf_v, srcI, dstI, m_enc,
                                                      buf_score, denom, buf_agg, Ee);
    attn_combine_kernel<<<nodeBlocks, 256, 0, stream>>>(buf_agg, denom, buf_s, buf_q, Nn);

    zero_stats_kernel<<<1, 256, 0, stream>>>(stats);
    colstats_kernel<Dd><<<statBlocks, Dd, 0, stream>>>(buf_q, colsum, colsumsq, Nn);
    bn_finalize_kernel<Dd><<<1, Dd, 0, stream>>>(colsum, colsumsq, bn_g + bofs,
                                                 bn_b + bofs, scale, shift, Nn);
    bn_apply_relu_kernel<Dd><<<nodeBlocks, 256, 0, stream>>>(buf_q, scale, shift,
                                                             buf_h, (long)nd);
    hin = buf_h;
  }

  // output head: [N,128] @ [128,64] + b, then BN + ReLU into d_out
  wmma_gemm_bias<OUTF><<<gemmGrid, 256, 0, stream>>>(buf_h, W_out, b_out, buf_k, Nn);
  zero_stats_kernel<<<1, 256, 0, stream>>>(stats);
  colstats_kernel<OUTF><<<statBlocks, OUTF, 0, stream>>>(buf_k, colsum, colsumsq, Nn);
  bn_finalize_kernel<OUTF><<<1, OUTF, 0, stream>>>(colsum, colsumsq, bno_g, bno_b,
                                                   scale, shift, Nn);
  const long totOut = (long)Nn * OUTF;
  bn_apply_relu_kernel<OUTF><<<(int)((totOut + 255) / 256), 256, 0, stream>>>(
      buf_k, scale, shift, (float*)d_out, totOut);
}